// Net3_59347858096347
// MI455X (gfx1250) — compile-verified
//
#include <hip/hip_runtime.h>
#include <math.h>

// ---------------------------------------------------------------------------
// Cosine-similarity argmax against a 100000x256 memory, MI455X (gfx1250).
// HBM-bound: 102.4 MB single stream -> ~4.4 us floor at 23.3 TB/s.
// Dot products via V_WMMA_F32_16X16X4_F32 (full f32 precision, matrix pipe),
// row norms on VALU in parallel, wave-level packed-key argmax + atomicMax.
// ---------------------------------------------------------------------------

typedef __attribute__((ext_vector_type(2))) float v2f;
typedef __attribute__((ext_vector_type(8))) float v8f;

static constexpr float  K_EPS  = 1e-8f;
static constexpr int    K_DIM  = 256;       // feature dimension
static constexpr int    K_ROWS_PER_WAVE = 16;
static constexpr int    K_WAVES_PER_BLK = 8;   // 256 threads = 8 wave32

// Map float -> monotonically ordered uint32 (total order matching float <)
__device__ __forceinline__ unsigned int float_to_ordered(float f) {
    unsigned int b = __float_as_uint(f);
    return (b & 0x80000000u) ? ~b : (b | 0x80000000u);
}
__device__ __forceinline__ float ordered_to_float(unsigned int u) {
    unsigned int b = (u & 0x80000000u) ? (u & 0x7FFFFFFFu) : ~u;
    return __uint_as_float(b);
}

// Kernel 1: zero the output vector and the argmax cell in workspace.
__global__ void zero_init_kernel(float* __restrict__ out, int n,
                                 unsigned long long* __restrict__ cell) {
    int i = blockIdx.x * blockDim.x + threadIdx.x;
    if (i < n) out[i] = 0.0f;
    if (i == 0) *cell = 0ULL;   // any ordered-encoded real float beats 0
}

// Kernel 2: one wave computes 16 rows' dot(x, row) via WMMA f32 16x16x4,
// row norms via VALU, then a packed (value, ~index) max reduction.
__global__ void __launch_bounds__(256)
cosine_argmax_wmma(const float* __restrict__ x,
                   const float* __restrict__ mem,
                   int nrows,
                   unsigned long long* __restrict__ cell) {
    const int lane = threadIdx.x & 31;
    const int wave = threadIdx.x >> 5;
    const int half = lane >> 4;                       // 0: K0..K1 slots, 1: K2..K3 slots
    const int tile = blockIdx.x * K_WAVES_PER_BLK + wave;
    const int base = tile * K_ROWS_PER_WAVE;
    if (base >= nrows) return;                        // whole-wave exit: EXEC all-1s for WMMA

    int row = base + (lane & 15);
    if (row >= nrows) row = nrows - 1;                // clamped dup: same idx and key as real row

    // lane reads float4 at columns 8*i + 4*half .. +3 of its row
    const float4* __restrict__ arow = (const float4*)(mem + (size_t)row * K_DIM) + half;
    const float4* __restrict__ xrow = (const float4*)x + half;

    v8f  c  = {};        // 16x16 accumulator: lane L, c[r] = dot(row base+r+8*(L>=16), x)
    float n2 = 0.0f;     // partial ||row(lane&15)||^2 over this half's columns

    #pragma unroll
    for (int i = 0; i < K_DIM / 8; ++i) {             // 32 iters, K advances 8/iter
        float4 F = arow[2 * i];
        float4 X = xrow[2 * i];
        v2f a0 = {F.x, F.y}, a1 = {F.z, F.w};
        v2f b0 = {X.x, X.y}, b1 = {X.z, X.w};
        // (neg_a, A, neg_b, B, c_mod, C, reuse_a, reuse_b)
        c = __builtin_amdgcn_wmma_f32_16x16x4_f32(false, a0, false, b0, (short)0, c, false, false);
        c = __builtin_amdgcn_wmma_f32_16x16x4_f32(false, a1, false, b1, (short)0, c, false, false);
        n2 = fmaf(F.x, F.x, n2);
        n2 = fmaf(F.y, F.y, n2);
        n2 = fmaf(F.z, F.z, n2);
        n2 = fmaf(F.w, F.w, n2);
    }

    // Combine the two half-row norm partials: lane m & m+16 both get full ||row base+m||^2
    n2 += __shfl_xor(n2, 16, 32);

    // Build packed candidates. argmax(dot/||row||) == argmax(cosine) since 1/||x||>0.
    unsigned long long best = 0ULL;
    #pragma unroll
    for (int r = 0; r < 8; ++r) {
        const int m = r + (half << 3);                // tile-row 0..15 this lane's c[r] holds
        float nr = __shfl(n2, m, 32);                 // ds_bpermute: norm^2 of row base+m
        float s  = c[r] / fmaxf(sqrtf(nr), K_EPS);
        int gidx = base + m;
        if (gidx >= nrows) gidx = nrows - 1;
        unsigned long long key =
            ((unsigned long long)float_to_ordered(s) << 32) |
            (unsigned long long)(0xFFFFFFFFu - (unsigned)gidx);  // ties -> smaller index
        if (key > best) best = key;
    }

    // Wave max-reduction of the 64-bit key (32-bit halves through shuffles)
    #pragma unroll
    for (int off = 16; off > 0; off >>= 1) {
        unsigned hi  = (unsigned)(best >> 32);
        unsigned lo  = (unsigned)best;
        unsigned ohi = __shfl_xor(hi, off, 32);
        unsigned olo = __shfl_xor(lo, off, 32);
        unsigned long long other = ((unsigned long long)ohi << 32) | olo;
        if (other > best) best = other;
    }
    if (lane == 0) atomicMax(cell, best);
}

// Kernel 3: decode winner, divide by ||x|| once, scatter single value.
__global__ void finalize_kernel(const float* __restrict__ x,
                                float* __restrict__ out,
                                const unsigned long long* __restrict__ cell,
                                int n) {
    if (blockIdx.x == 0 && threadIdx.x == 0) {
        unsigned long long key = *cell;
        unsigned u  = (unsigned)(key >> 32);
        int idx     = (int)(0xFFFFFFFFu - (unsigned)(key & 0xFFFFFFFFu));
        float s     = ordered_to_float(u);
        float xn2 = 0.0f;
        for (int i = 0; i < K_DIM; ++i) xn2 = fmaf(x[i], x[i], xn2);
        float xn = fmaxf(sqrtf(xn2), K_EPS);
        if (idx >= 0 && idx < n) out[idx] = s / xn;
    }
}

extern "C" void kernel_launch(void* const* d_in, const int* in_sizes, int n_in,
                              void* d_out, int out_size, void* d_ws, size_t ws_size,
                              hipStream_t stream) {
    const float* x   = (const float*)d_in[0];     // [256]
    const float* mem = (const float*)d_in[1];     // [100000, 256]
    float*       out = (float*)d_out;             // [100000]
    const int nrows  = in_sizes[1] / K_DIM;

    unsigned long long* cell = (unsigned long long*)d_ws;

    // 1) zero output + argmax cell (d_out / d_ws are poisoned)
    int zblocks = (out_size + 255) / 256;
    zero_init_kernel<<<zblocks, 256, 0, stream>>>(out, out_size, cell);

    // 2) main WMMA GEMV + argmax: 16 rows/wave, 8 waves/block
    int tiles  = (nrows + K_ROWS_PER_WAVE - 1) / K_ROWS_PER_WAVE;
    int blocks = (tiles + K_WAVES_PER_BLK - 1) / K_WAVES_PER_BLK;
    cosine_argmax_wmma<<<blocks, 256, 0, stream>>>(x, mem, nrows, cell);

    // 3) decode + scatter single winner
    finalize_kernel<<<1, 64, 0, stream>>>(x, out, cell, out_size);
}